// CognitiveManifold_51410758533316
// MI455X (gfx1250) — compile-verified
//
#include <hip/hip_runtime.h>
#include <math.h>

// CDNA5 / gfx1250 wave32 WMMA geodesic-RK2 kernel.
// 1 wave = 16 points; all MLP matmuls via V_WMMA_F32_16X16X4_F32.

typedef float v8f __attribute__((ext_vector_type(8)));
typedef float v2f __attribute__((ext_vector_type(2)));

#define FD_EPS_F 1e-4f
#define GDT 0.1f

struct SM {
  // weights
  float W1[128];   // [8][16]
  float b1[16];
  float W2[1024];  // [16][64]
  float b2[64];
  float Wr1[64];   // [8][8]
  float br1[8];
  float Wr2[8];
  float br2;
  float g0[64];    // L @ L^T
  float Ltmp[64];
  // per-tile state (16 points)
  float x0[128], v0[128];   // [16][8]
  float xs[128], vs[128];   // stage inputs
  float abuf[128];          // stage output accel
  // metric-eval scratch
  float hbuf[256];          // [16][16] hidden
  float h2buf[1024];        // [16][64] raw h
  float ubuf[128];          // [16][8] r-branch hidden
  float rbuf[16];           // r^2 per point
  // metric buffers
  float gc[1024];           // center metric [16][8][8]
  float gp[1024];           // +eps
  float gm[1024];           // -eps
  float ginv[1024];         // inverse of center metric
};

__device__ __forceinline__ float gelu_exact(float x) {
  return 0.5f * x * (1.0f + erff(x * 0.70710678118654752f));
}
__device__ __forceinline__ float softplus_f(float x) {
  return (x > 0.f) ? (x + log1pf(expf(-x))) : log1pf(expf(x));
}

// Evaluate tilde_g(xsrc + s*e_rho) for 16 points -> outg[16][64].
// All 32 lanes must be active (WMMA requires EXEC == all-1s).
__device__ __attribute__((noinline)) void metric_eval(SM* sm, const float* xsrc,
                                                      int rho, float s, float* outg) {
  const int lane = threadIdx.x;      // blockDim.x == 32
  const int m  = lane & 15;          // point row (A) / column (B,C)
  const int hi = lane >> 4;          // K-half / M-half selector
  const int ln = m;
  const int k0 = 2 * hi;             // A/B K indices for K-step 0
  const int k1 = 4 + 2 * hi;         // for K-step 1

  // ---- A operand: X = xsrc (+ s on coordinate rho), 16x8, two K=4 chunks
  v2f a0, a1, b;
  a0.x = xsrc[m * 8 + k0]     + ((k0     == rho) ? s : 0.f);
  a0.y = xsrc[m * 8 + k0 + 1] + ((k0 + 1 == rho) ? s : 0.f);
  a1.x = xsrc[m * 8 + k1]     + ((k1     == rho) ? s : 0.f);
  a1.y = xsrc[m * 8 + k1 + 1] + ((k1 + 1 == rho) ? s : 0.f);

  // ---- hidden = gelu(X @ W1 + b1): 16 pts x 16 hidden, K=8 -> 2 WMMA
  v8f acc = {};
  b.x = sm->W1[k0 * 16 + ln]; b.y = sm->W1[(k0 + 1) * 16 + ln];
  acc = __builtin_amdgcn_wmma_f32_16x16x4_f32(false, a0, false, b, (short)0, acc, false, false);
  b.x = sm->W1[k1 * 16 + ln]; b.y = sm->W1[(k1 + 1) * 16 + ln];
  acc = __builtin_amdgcn_wmma_f32_16x16x4_f32(false, a1, false, b, (short)0, acc, false, false);
#pragma unroll
  for (int j = 0; j < 8; ++j)
    sm->hbuf[(j + 8 * hi) * 16 + ln] = gelu_exact(acc[j] + sm->b1[ln]);

  // ---- r-branch: u = gelu(X @ Wr1 + br1)  (N=8 padded to 16)
  v8f accr = {};
  b.x = (ln < 8) ? sm->Wr1[k0 * 8 + ln] : 0.f;
  b.y = (ln < 8) ? sm->Wr1[(k0 + 1) * 8 + ln] : 0.f;
  accr = __builtin_amdgcn_wmma_f32_16x16x4_f32(false, a0, false, b, (short)0, accr, false, false);
  b.x = (ln < 8) ? sm->Wr1[k1 * 8 + ln] : 0.f;
  b.y = (ln < 8) ? sm->Wr1[(k1 + 1) * 8 + ln] : 0.f;
  accr = __builtin_amdgcn_wmma_f32_16x16x4_f32(false, a1, false, b, (short)0, accr, false, false);
  if (ln < 8) {
#pragma unroll
    for (int j = 0; j < 8; ++j)
      sm->ubuf[(j + 8 * hi) * 8 + ln] = gelu_exact(accr[j] + sm->br1[ln]);
  }
  __syncthreads();

  // ---- h2 = hidden @ W2 + b2: 16 pts x 64, K=16 -> 4 N-tiles x 4 K-steps
  v2f ha[4];
#pragma unroll
  for (int kk = 0; kk < 4; ++kk) {
    ha[kk].x = sm->hbuf[m * 16 + 4 * kk + 2 * hi];
    ha[kk].y = sm->hbuf[m * 16 + 4 * kk + 2 * hi + 1];
  }
#pragma unroll
  for (int c = 0; c < 4; ++c) {
    v8f acc2 = {};
#pragma unroll
    for (int kk = 0; kk < 4; ++kk) {
      v2f bb;
      bb.x = sm->W2[(4 * kk + 2 * hi) * 64 + 16 * c + ln];
      bb.y = sm->W2[(4 * kk + 2 * hi + 1) * 64 + 16 * c + ln];
      acc2 = __builtin_amdgcn_wmma_f32_16x16x4_f32(false, ha[kk], false, bb, (short)0, acc2, false, false);
    }
#pragma unroll
    for (int j = 0; j < 8; ++j)
      sm->h2buf[(j + 8 * hi) * 64 + 16 * c + ln] = acc2[j] + sm->b2[16 * c + ln];
  }
  __syncthreads();

  // ---- r^2 per point (lanes 0..15)
  if (lane < 16) {
    const int p = lane;
    float rr = sm->br2;
#pragma unroll
    for (int k = 0; k < 8; ++k) rr += sm->ubuf[p * 8 + k] * sm->Wr2[k];
    rr = softplus_f(rr);
    rr = fminf(fmaxf(rr, 0.1f), 10.0f);
    sm->rbuf[p] = rr * rr;
  }
  __syncthreads();

  // ---- assembly on all 32 lanes: 1024 entries, 32 per lane
  for (int it = 0; it < 32; ++it) {
    const int idx = it * 32 + lane;
    const int p = idx >> 6;
    const int ij = idx & 63;
    const int i = ij >> 3, jj = ij & 7;
    float sym = 0.5f * (sm->h2buf[p * 64 + ij] + sm->h2buf[p * 64 + jj * 8 + i]);
    float val = sm->g0[ij] + 0.1f * tanhf(sym) + ((i == jj) ? 1e-4f : 0.f);
    outg[idx] = sm->rbuf[p] * val;
  }
  __syncthreads();
}

// a^l = -Gamma^l_{mn} v^m v^n for the 16 points in sm->xs / sm->vs -> sm->abuf.
__device__ __attribute__((noinline)) void christoffel_accel(SM* sm) {
  const int lane = threadIdx.x;

  // center metric + per-point 8x8 inverse (SPD -> Gauss-Jordan, no pivoting)
  metric_eval(sm, sm->xs, -1, 0.f, sm->gc);
  if (lane < 16) {
    float A[64];
#pragma unroll
    for (int i = 0; i < 64; ++i) A[i] = sm->gc[lane * 64 + i];
    for (int p = 0; p < 8; ++p) {
      float ipiv = 1.0f / A[p * 8 + p];
      A[p * 8 + p] = 1.0f;
      for (int j = 0; j < 8; ++j) A[p * 8 + j] *= ipiv;
      for (int i2 = 0; i2 < 8; ++i2) {
        if (i2 == p) continue;
        float f = A[i2 * 8 + p];
        A[i2 * 8 + p] = 0.f;
        for (int j = 0; j < 8; ++j) A[i2 * 8 + j] -= f * A[p * 8 + j];
      }
    }
#pragma unroll
    for (int i = 0; i < 64; ++i) sm->ginv[lane * 64 + i] = A[i];
  }
  __syncthreads();

  // Using symmetry of g: a^l = -g^{lr} (c_r - 0.5 e_r),
  //   c_r = sum_rho v_rho (dg[rho] v)_r,   e_rho = v^T dg[rho] v.
  float cacc[8], eacc[8];
#pragma unroll
  for (int r = 0; r < 8; ++r) { cacc[r] = 0.f; eacc[r] = 0.f; }

#pragma unroll
  for (int rho = 0; rho < 8; ++rho) {
    metric_eval(sm, sm->xs, rho,  FD_EPS_F, sm->gp);
    metric_eval(sm, sm->xs, rho, -FD_EPS_F, sm->gm);
    if (lane < 16) {
      const int p = lane;
      float vv[8];
#pragma unroll
      for (int n = 0; n < 8; ++n) vv[n] = sm->vs[p * 8 + n];
      const float inv2e = 1.0f / (2.0f * FD_EPS_F);
      const float vr = vv[rho];
      float q = 0.f;
#pragma unroll
      for (int r = 0; r < 8; ++r) {
        float t = 0.f;
#pragma unroll
        for (int n = 0; n < 8; ++n)
          t += (sm->gp[p * 64 + r * 8 + n] - sm->gm[p * 64 + r * 8 + n]) * vv[n];
        t *= inv2e;
        cacc[r] += vr * t;
        q += vv[r] * t;
      }
      eacc[rho] = q;
    }
    __syncthreads();
  }

  if (lane < 16) {
    const int p = lane;
    for (int l = 0; l < 8; ++l) {
      float a = 0.f;
#pragma unroll
      for (int r = 0; r < 8; ++r)
        a -= sm->ginv[p * 64 + l * 8 + r] * (cacc[r] - 0.5f * eacc[r]);
      sm->abuf[p * 8 + l] = a;
    }
  }
  __syncthreads();
}

__global__ __launch_bounds__(32) void geodesic_rk2_kernel(
    const float* __restrict__ x, const float* __restrict__ v,
    const float* __restrict__ L, const float* __restrict__ W1, const float* __restrict__ b1,
    const float* __restrict__ W2, const float* __restrict__ b2,
    const float* __restrict__ Wr1, const float* __restrict__ br1,
    const float* __restrict__ Wr2, const float* __restrict__ br2,
    float* __restrict__ out, long long nelem) {
  __shared__ SM sm;
  const int tid = threadIdx.x;

  // preload weights
  for (int i = tid; i < 128; i += 32) sm.W1[i] = W1[i];
  for (int i = tid; i < 1024; i += 32) sm.W2[i] = W2[i];
  for (int i = tid; i < 64; i += 32) { sm.b2[i] = b2[i]; sm.Wr1[i] = Wr1[i]; sm.Ltmp[i] = L[i]; }
  if (tid < 16) sm.b1[tid] = b1[tid];
  if (tid < 8) { sm.br1[tid] = br1[tid]; sm.Wr2[tid] = Wr2[tid]; }
  if (tid == 0) sm.br2 = br2[0];

  // load 16-point tile of x, v
  const long long pbase = (long long)blockIdx.x * 16;
  for (int i = tid; i < 128; i += 32) {
    long long g = pbase * 8 + i;
    float xv = (g < nelem) ? x[g] : 0.f;
    float vvv = (g < nelem) ? v[g] : 0.f;
    sm.x0[i] = xv; sm.v0[i] = vvv;
    sm.xs[i] = xv; sm.vs[i] = vvv;
  }
  __syncthreads();

  // g0 = L @ L^T
  for (int idx = tid; idx < 64; idx += 32) {
    int i = idx >> 3, j = idx & 7;
    float s = 0.f;
    for (int k = 0; k < 8; ++k) s += sm.Ltmp[i * 8 + k] * sm.Ltmp[j * 8 + k];
    sm.g0[idx] = s;
  }
  __syncthreads();

  // RK2 stage 1: a = accel(x0, v0)
  christoffel_accel(&sm);

  // midpoint state; x_new = x0 + dt * v_mid can be emitted now
  for (int i = tid; i < 128; i += 32) {
    float vm = sm.v0[i] + 0.5f * GDT * sm.abuf[i];
    sm.xs[i] = sm.x0[i] + 0.5f * GDT * sm.v0[i];
    sm.vs[i] = vm;
    long long g = pbase * 8 + i;
    if (g < nelem) out[g] = sm.x0[i] + GDT * vm;
  }
  __syncthreads();

  // RK2 stage 2: a_mid = accel(x_mid, v_mid)
  christoffel_accel(&sm);

  for (int i = tid; i < 128; i += 32) {
    long long g = pbase * 8 + i;
    if (g < nelem) out[nelem + g] = sm.v0[i] + GDT * sm.abuf[i];
  }
}

extern "C" void kernel_launch(void* const* d_in, const int* in_sizes, int n_in,
                              void* d_out, int out_size, void* d_ws, size_t ws_size,
                              hipStream_t stream) {
  const float* x   = (const float*)d_in[0];
  const float* v   = (const float*)d_in[1];
  const float* L   = (const float*)d_in[2];
  const float* W1  = (const float*)d_in[3];
  const float* b1  = (const float*)d_in[4];
  const float* W2  = (const float*)d_in[5];
  const float* b2  = (const float*)d_in[6];
  const float* Wr1 = (const float*)d_in[7];
  const float* br1 = (const float*)d_in[8];
  const float* Wr2 = (const float*)d_in[9];
  const float* br2 = (const float*)d_in[10];
  float* out = (float*)d_out;

  long long nelem = (long long)in_sizes[0];   // B*N*D
  long long npts  = nelem / 8;
  int nblocks = (int)((npts + 15) / 16);

  geodesic_rk2_kernel<<<nblocks, 32, 0, stream>>>(
      x, v, L, W1, b1, W2, b2, Wr1, br1, Wr2, br2, out, nelem);
}